// QueueMemory_29033978921655
// MI455X (gfx1250) — compile-verified
//
#include <hip/hip_runtime.h>

typedef __attribute__((ext_vector_type(2))) float v2f;
typedef __attribute__((ext_vector_type(8))) float v8f;

namespace {
constexpr int   kT   = 128;
constexpr int   kF   = 512;
constexpr int   kLen = 65536;
constexpr float kEps = 0.51f;

// d_ws layout (32-bit words):
//  0: s_norm2 (f)   1: reward_sum (f)
//  2: cq_max (f)    3: mi_q (i)    4: cond_q (i)
//  5: cs_max (f)    6: mi_s (i)    7: cond_s (i)
//  8: r (i)        10: out_i (i)
constexpr int kCpv = 16;            // 512 compat partial max values
constexpr int kCpi = kCpv + 512;    // 512 compat partial argmax
constexpr int kCpf = kCpi + 512;    // 512 compat partial any-flags
constexpr int kIpv = kCpf + 512;    // 64 idx-scan partial min values
constexpr int kIpi = kIpv + 64;     // 64 idx-scan partial argmin
constexpr int kJpv = kIpi + 64;     // 64 idx2-scan partial max values
constexpr int kJpi = kJpv + 64;     // 64 idx2-scan partial argmax
}  // namespace

// ---- Kernel A: ||s||^2 and reward_sum ------------------------------------
__global__ void qm_prep(const float* __restrict__ seq, float* __restrict__ wsf) {
  __shared__ float red[kF];
  const int t = threadIdx.x;
  const float sv = seq[(kT - 1) * kF + t];
  red[t] = sv * sv;
  __syncthreads();
  for (int s = kF / 2; s > 0; s >>= 1) {
    if (t < s) red[t] += red[t + s];
    __syncthreads();
  }
  if (t == 0) wsf[0] = red[0];
  __syncthreads();
  red[t] = (t < kT) ? seq[t * kF + (kF - 1)] : 0.0f;
  __syncthreads();
  for (int s = kF / 2; s > 0; s >>= 1) {
    if (t < s) red[t] += red[t + s];
    __syncthreads();
  }
  if (t == 0) wsf[1] = red[0];
}

// ---- Kernel B: compat scan over memory (blocks 0..255) & route (256..511) -
// Each block: 128 threads = 4 waves; each wave owns a 16-row WMMA tile;
// block covers 256 rows in 4 chunk iterations.
__global__ void __launch_bounds__(128)
qm_compat_scan(const float* __restrict__ seq, const float* __restrict__ mem,
               const float* __restrict__ route, float* __restrict__ wsf,
               int* __restrict__ wsi) {
  __shared__ float sS[kF];        // broadcast states vector
  __shared__ float sDot[4 * 16];  // per-wave dot extraction scratch
  __shared__ float rv[128];
  __shared__ int   ri[128];
  __shared__ int   rf[128];

  const float* __restrict__ states = seq + (kT - 1) * kF;
  const int arr = blockIdx.x >> 8;   // 0 = memory, 1 = maximum_route
  const int blk = blockIdx.x & 255;
  const float* __restrict__ src = arr ? route : mem;

  for (int i = threadIdx.x; i < kF; i += blockDim.x) sS[i] = states[i];
  __syncthreads();
  const float snorm2 = wsf[0];

  const int tid  = threadIdx.x;
  const int lane = tid & 31;
  const int wid  = tid >> 5;
  const int m    = lane & 15;   // tile row owned by this lane (A-matrix M)
  const int half = lane >> 4;   // K-pair select: {0,1} vs {2,3}

  float bestv = -3.0e30f;
  int   besti = 0;
  int   anyf  = 0;

  for (int chunk = 0; chunk < 4; ++chunk) {
    const int row0 = blk * 256 + chunk * 64 + wid * 16;
    const float* base = src + (size_t)(row0 + m) * kF + 2 * half;
    __builtin_prefetch(base + 64 * kF, 0, 1);  // next chunk's rows

    v8f   acc  = {0.f, 0.f, 0.f, 0.f, 0.f, 0.f, 0.f, 0.f};
    float nacc = 0.f;
    for (int kb = 0; kb < kF; kb += 4) {
      // A 16x4 fp32 fragment: lane holds row m, K = 2*half + {0,1}
      const v2f a = *reinterpret_cast<const v2f*>(base + kb);
      // B 4x16 fp32: all 16 N-columns = state chunk (broadcast)
      v2f b;
      b.x = sS[kb + 2 * half];
      b.y = sS[kb + 2 * half + 1];
      acc = __builtin_amdgcn_wmma_f32_16x16x4_f32(false, a, false, b, (short)0,
                                                  acc, false, false);
      nacc = fmaf(a.x, a.x, nacc);
      nacc = fmaf(a.y, a.y, nacc);
    }
    // full ||row m||^2: combine K halves held by lane m and lane m+16
    nacc += __shfl_xor(nacc, 16, 32);

    // D[j][n] (all n identical): lanes 0-15 -> rows j, lanes 16-31 -> rows j+8
    if (m == 0) {
#pragma unroll
      for (int j = 0; j < 8; ++j) sDot[wid * 16 + j + 8 * half] = acc[j];
    }
    __syncthreads();
    if (lane < 16) {
      const float dot = sDot[wid * 16 + m];
      const float lv2 = fmaxf(nacc - 2.0f * dot + snorm2, 0.0f);
      const float lv  = sqrtf(lv2);
      const float hs  = fminf(fmaxf(fmaf(0.2f, lv, 0.5f), 0.0f), 1.0f);
      const float c   = 0.5f - hs;
      const int rowi  = row0 + m;
      anyf |= (c >= kEps) ? 1 : 0;
      if (c > bestv || (c == bestv && rowi < besti)) { bestv = c; besti = rowi; }
    }
    __syncthreads();
  }

  rv[tid] = bestv; ri[tid] = besti; rf[tid] = anyf;
  __syncthreads();
  for (int s = 64; s > 0; s >>= 1) {
    if (tid < s) {
      const float v2 = rv[tid + s];
      const int   i2 = ri[tid + s];
      if (v2 > rv[tid] || (v2 == rv[tid] && i2 < ri[tid])) { rv[tid] = v2; ri[tid] = i2; }
      rf[tid] |= rf[tid + s];
    }
    __syncthreads();
  }
  if (tid == 0) {
    wsf[kCpv + blockIdx.x] = rv[0];
    wsi[kCpi + blockIdx.x] = ri[0];
    wsi[kCpf + blockIdx.x] = rf[0];
  }
}

// ---- Kernel C: reduce compat partials (block 0 -> q scalars, 1 -> s) ------
__global__ void qm_compat_reduce(float* __restrict__ wsf, int* __restrict__ wsi) {
  __shared__ float rv[256];
  __shared__ int   ri[256];
  __shared__ int   rf[256];
  const int a = blockIdx.x;
  const int t = threadIdx.x;
  rv[t] = wsf[kCpv + a * 256 + t];
  ri[t] = wsi[kCpi + a * 256 + t];
  rf[t] = wsi[kCpf + a * 256 + t];
  __syncthreads();
  for (int s = 128; s > 0; s >>= 1) {
    if (t < s) {
      const float v2 = rv[t + s];
      const int   i2 = ri[t + s];
      if (v2 > rv[t] || (v2 == rv[t] && i2 < ri[t])) { rv[t] = v2; ri[t] = i2; }
      rf[t] |= rf[t + s];
    }
    __syncthreads();
  }
  if (t == 0) {
    if (a == 0) { wsf[2] = rv[0]; wsi[3] = ri[0]; wsi[4] = rf[0]; }
    else        { wsf[5] = rv[0]; wsi[6] = ri[0]; wsi[7] = rf[0]; }
  }
}

// ---- Kernel D: argmin over idx1 (index with mi_q substituted if cond_q) ---
__global__ void qm_idx_scan(const float* __restrict__ index,
                            float* __restrict__ wsf, int* __restrict__ wsi) {
  __shared__ float rv[256];
  __shared__ int   ri[256];
  const int   cond_q = wsi[4];
  const int   mi_q   = wsi[3];
  const float cqm    = wsf[2];
  const float rsum   = wsf[1];
  const float upq    = index[mi_q] * (1.0f - cqm) + rsum * cqm;

  float bv = 3.0e38f;
  int   bi = 0;
  for (int k = 0; k < 4; ++k) {
    const int i = blockIdx.x * 1024 + k * 256 + threadIdx.x;
    float v = index[i];
    if (cond_q && i == mi_q) v = upq;
    if (v < bv || (v == bv && i < bi)) { bv = v; bi = i; }
  }
  const int t = threadIdx.x;
  rv[t] = bv; ri[t] = bi;
  __syncthreads();
  for (int s = 128; s > 0; s >>= 1) {
    if (t < s) {
      const float v2 = rv[t + s];
      const int   i2 = ri[t + s];
      if (v2 < rv[t] || (v2 == rv[t] && i2 < ri[t])) { rv[t] = v2; ri[t] = i2; }
    }
    __syncthreads();
  }
  if (t == 0) { wsf[kIpv + blockIdx.x] = rv[0]; wsi[kIpi + blockIdx.x] = ri[0]; }
}

__global__ void qm_r_reduce(float* __restrict__ wsf, int* __restrict__ wsi) {
  __shared__ float rv[64];
  __shared__ int   ri[64];
  const int t = threadIdx.x;
  rv[t] = wsf[kIpv + t];
  ri[t] = wsi[kIpi + t];
  __syncthreads();
  for (int s = 32; s > 0; s >>= 1) {
    if (t < s) {
      const float v2 = rv[t + s];
      const int   i2 = ri[t + s];
      if (v2 < rv[t] || (v2 == rv[t] && i2 < ri[t])) { rv[t] = v2; ri[t] = i2; }
    }
    __syncthreads();
  }
  if (t == 0) wsi[8] = ri[0];
}

// ---- Kernel F: argmax over idx2 (remove row r, append app_idx) ------------
__global__ void qm_idx2_scan(const float* __restrict__ index,
                             const float* __restrict__ route,
                             float* __restrict__ wsf, int* __restrict__ wsi) {
  __shared__ float rv[256];
  __shared__ int   ri[256];
  const int   cond_q = wsi[4], mi_q = wsi[3];
  const int   cond_s = wsi[7], mi_s = wsi[6];
  const int   r      = wsi[8];
  const float cqm = wsf[2], csm = wsf[5], rsum = wsf[1];
  const float upq = index[mi_q] * (1.0f - cqm) + rsum * cqm;
  const int   do_append = cond_s | (cond_q ? 0 : 1);
  float app_idx = rsum;
  if (cond_s) app_idx = route[(size_t)mi_s * kF + (kF - 1)] * csm + rsum * (1.0f - csm);

  float bv = -3.0e38f;
  int   bi = 0;
  for (int k = 0; k < 4; ++k) {
    const int i = blockIdx.x * 1024 + k * 256 + threadIdx.x;
    float v;
    if (!do_append) {
      v = (cond_q && i == mi_q) ? upq : index[i];
    } else if (i == kLen - 1) {
      v = app_idx;
    } else {
      const int j = (i < r) ? i : i + 1;
      v = (cond_q && j == mi_q) ? upq : index[j];
    }
    if (v > bv || (v == bv && i < bi)) { bv = v; bi = i; }
  }
  const int t = threadIdx.x;
  rv[t] = bv; ri[t] = bi;
  __syncthreads();
  for (int s = 128; s > 0; s >>= 1) {
    if (t < s) {
      const float v2 = rv[t + s];
      const int   i2 = ri[t + s];
      if (v2 > rv[t] || (v2 == rv[t] && i2 < ri[t])) { rv[t] = v2; ri[t] = i2; }
    }
    __syncthreads();
  }
  if (t == 0) { wsf[kJpv + blockIdx.x] = rv[0]; wsi[kJpi + blockIdx.x] = ri[0]; }
}

__global__ void qm_outi_reduce(float* __restrict__ wsf, int* __restrict__ wsi,
                               float* __restrict__ out) {
  __shared__ float rv[64];
  __shared__ int   ri[64];
  const int t = threadIdx.x;
  rv[t] = wsf[kJpv + t];
  ri[t] = wsi[kJpi + t];
  __syncthreads();
  for (int s = 32; s > 0; s >>= 1) {
    if (t < s) {
      const float v2 = rv[t + s];
      const int   i2 = ri[t + s];
      if (v2 > rv[t] || (v2 == rv[t] && i2 < ri[t])) { rv[t] = v2; ri[t] = i2; }
    }
    __syncthreads();
  }
  if (t == 0) {
    wsi[10]  = ri[0];
    out[kF]  = rv[0];  // idx_out = idx2[out_i]
  }
}

// ---- Kernel H: gather output row mem2[out_i] ------------------------------
__global__ void qm_emit(const float* __restrict__ seq,
                        const float* __restrict__ route,
                        const float* __restrict__ mem,
                        const float* __restrict__ wsf, const int* __restrict__ wsi,
                        float* __restrict__ out) {
  const int f = threadIdx.x;
  const int cond_q = wsi[4], mi_q = wsi[3];
  const int cond_s = wsi[7], mi_s = wsi[6];
  const int r = wsi[8], oi = wsi[10];
  const float cqm = wsf[2], csm = wsf[5];
  const float sf = seq[(kT - 1) * kF + f];
  const int do_append = cond_s | (cond_q ? 0 : 1);
  float v;
  if (do_append && oi == kLen - 1) {
    v = cond_s ? (route[(size_t)mi_s * kF + f] * csm + sf * (1.0f - csm)) : sf;
  } else {
    const int j = do_append ? ((oi < r) ? oi : oi + 1) : oi;
    if (cond_q && j == mi_q) {
      v = mem[(size_t)mi_q * kF + f] * (1.0f - cqm) + sf * cqm;
    } else {
      v = mem[(size_t)j * kF + f];
    }
  }
  out[f] = v;
}

extern "C" void kernel_launch(void* const* d_in, const int* in_sizes, int n_in,
                              void* d_out, int out_size, void* d_ws, size_t ws_size,
                              hipStream_t stream) {
  const float* seq   = (const float*)d_in[0];  // states_seq  (1,128,512)
  const float* route = (const float*)d_in[1];  // maximum_route (1,65536,512)
  const float* mem   = (const float*)d_in[2];  // memory      (1,65536,512)
  const float* index = (const float*)d_in[3];  // index       (1,65536,1)
  float* out = (float*)d_out;                  // 512 mem_out + 1 idx_out
  float* wsf = (float*)d_ws;
  int*   wsi = (int*)d_ws;

  qm_prep<<<1, 512, 0, stream>>>(seq, wsf);
  qm_compat_scan<<<512, 128, 0, stream>>>(seq, mem, route, wsf, wsi);
  qm_compat_reduce<<<2, 256, 0, stream>>>(wsf, wsi);
  qm_idx_scan<<<64, 256, 0, stream>>>(index, wsf, wsi);
  qm_r_reduce<<<1, 64, 0, stream>>>(wsf, wsi);
  qm_idx2_scan<<<64, 256, 0, stream>>>(index, route, wsf, wsi);
  qm_outi_reduce<<<1, 64, 0, stream>>>(wsf, wsi, out);
  qm_emit<<<1, 512, 0, stream>>>(seq, route, mem, wsf, wsi, out);
}